// Gather3d_52905407152580
// MI455X (gfx1250) — compile-verified
//
#include <hip/hip_runtime.h>

// Gather3d for x:[1,128,16,128,128] f32, active_indices:[128,2] i32
// out:[128,128,18,16,16] f32
//   out[n,c,tt,i,j] = (tt < 2) ? 0 : x[0, c, tt-2, idx0[n]+i, idx1[n]+j]
//
// Pure memory op (0 FLOPs): ~302MB NT stores + reads that stay resident in the
// 192MB L2 (x is 128MB). Roofline @23.3TB/s ~= 25us. All address math is
// 32-bit shift/mask (no division); block = one (n,c) pair.

typedef float f4 __attribute__((ext_vector_type(4)));

#define C_   128u
#define T_   16u
#define HW_  16384u   // 128*128 plane of x
#define TT_  18u      // T_ + PAD_T(2)

__global__ __launch_bounds__(192) void gather3d_copy_kernel(
    const float* __restrict__ x,
    const int*   __restrict__ idx,
    float*       __restrict__ out)
{
    unsigned nc  = blockIdx.x;        // n*128 + c   (16384 blocks)
    unsigned n   = nc >> 7;
    unsigned c   = nc & 127u;

    unsigned tid   = threadIdx.x;     // 192 threads = 6 wave32
    unsigned pslot = tid >> 6;        // 0..2 : plane slot (uniform per 2 waves)
    unsigned v     = tid & 63u;       // float4 slot within a 256-float plane
    unsigned i     = v >> 2;          // tile row 0..15
    unsigned jq    = v & 3u;          // float4 within 16-float row

    // Block-uniform index pair -> scalar loads
    int y0 = idx[2u * n + 0u];
    int x0 = idx[2u * n + 1u];

    // All offsets fit comfortably in 32 bits (out: 302MB, x: 134MB)
    float*       ob = out + nc * (TT_ * 256u) + v * 4u;
    const float* ib = x + c * (T_ * HW_) + (unsigned)(y0 + (int)i) * 128u
                        + (unsigned)x0 + jq * 4u;

    // 18 planes = 6 iterations x 3 slots; full unroll -> load clause in flight
#pragma unroll
    for (int it = 0; it < 6; ++it) {
        int tt = it * 3 + (int)pslot;
        f4 val;
        if (tt < 2) {
            // causal time padding: two zero frames (wave-uniform branch)
            val = (f4){0.0f, 0.0f, 0.0f, 0.0f};
        } else {
            const float* ip = ib + (unsigned)(tt - 2) * HW_;
            val.x = ip[0];
            val.y = ip[1];
            val.z = ip[2];
            val.w = ip[3];
        }
        // write-once output: NT store, don't evict the L2-resident input
        __builtin_nontemporal_store(val, (f4*)(ob + (unsigned)tt * 256u));
    }
}

extern "C" void kernel_launch(void* const* d_in, const int* in_sizes, int n_in,
                              void* d_out, int out_size, void* d_ws, size_t ws_size,
                              hipStream_t stream) {
    const float* x   = (const float*)d_in[0];   // 1*128*16*128*128 f32
    const int*   idx = (const int*)  d_in[1];   // 128*2 i32
    float*       out = (float*)d_out;           // 128*128*18*16*16 f32

    const unsigned blocks  = 128u * 128u;       // one block per (n, c)
    const unsigned threads = 192u;              // 3 plane-slots x 64 threads

    gather3d_copy_kernel<<<blocks, threads, 0, stream>>>(x, idx, out);
}